// GCN_CFLP_60988535603566
// MI455X (gfx1250) — compile-verified
//
#include <hip/hip_runtime.h>
#include <hip/hip_bf16.h>

typedef __attribute__((ext_vector_type(2))) float v2f;
typedef __attribute__((ext_vector_type(8))) float v8f;
typedef int v4i __attribute__((vector_size(16)));

#define FDIM 128   // input feature width for every layer (F_IN = H = 128)

#define USE_ASYNC_STAGE 1
#if USE_ASYNC_STAGE && __has_builtin(__builtin_amdgcn_global_load_async_to_lds_b128)
#define HAVE_ASYNC 1
#else
#define HAVE_ASYNC 0
#endif

// ---------------------------------------------------------------------------
// Degree kernels: integer counts (init 1 == self-loop), later rsqrt'ed.
// ---------------------------------------------------------------------------
__global__ void deg_init(unsigned* __restrict__ od, unsigned* __restrict__ id, int N) {
    int i = blockIdx.x * blockDim.x + threadIdx.x;
    if (i < N) { od[i] = 1u; id[i] = 1u; }
}

__global__ void deg_acc(const int* __restrict__ src, const int* __restrict__ dst,
                        unsigned* __restrict__ od, unsigned* __restrict__ id, int E) {
    int e = blockIdx.x * blockDim.x + threadIdx.x;
    if (e < E) {
        atomicAdd(&od[src[e]], 1u);
        atomicAdd(&id[dst[e]], 1u);
    }
}

__global__ void deg_fin(float* __restrict__ dout, float* __restrict__ din, int N) {
    int i = blockIdx.x * blockDim.x + threadIdx.x;
    if (i < N) {
        unsigned o = ((const unsigned*)dout)[i];
        unsigned d = ((const unsigned*)din)[i];
        dout[i] = rsqrtf((float)o);
        din[i]  = rsqrtf((float)d);
    }
}

// ---------------------------------------------------------------------------
// Single-block chunked exclusive scan of real in-degrees (cnt[i]-1) -> row_ptr.
// ---------------------------------------------------------------------------
#define SCAN_T 1024
__global__ void scan_rowptr(const unsigned* __restrict__ cnt, unsigned* __restrict__ row_ptr,
                            int N) {
    __shared__ unsigned buf[SCAN_T];
    __shared__ unsigned carry;
    if (threadIdx.x == 0) carry = 0u;
    __syncthreads();
    for (int base = 0; base < N; base += SCAN_T) {
        int i = base + (int)threadIdx.x;
        unsigned v = (i < N) ? (cnt[i] - 1u) : 0u;   // counts include +1 self-loop init
        buf[threadIdx.x] = v;
        __syncthreads();
        for (int off = 1; off < SCAN_T; off <<= 1) {
            unsigned t = (threadIdx.x >= (unsigned)off) ? buf[threadIdx.x - off] : 0u;
            __syncthreads();
            buf[threadIdx.x] += t;
            __syncthreads();
        }
        if (i < N) row_ptr[i] = carry + buf[threadIdx.x] - v;  // exclusive
        unsigned total = buf[SCAN_T - 1];
        __syncthreads();
        if (threadIdx.x == 0) carry += total;
        __syncthreads();
    }
    if (threadIdx.x == 0) row_ptr[N] = carry;
}

__global__ void copy_u32(const unsigned* __restrict__ a, unsigned* __restrict__ b, int n) {
    int i = blockIdx.x * blockDim.x + threadIdx.x;
    if (i < n) b[i] = a[i];
}

__global__ void fill_edges(const int* __restrict__ src, const int* __restrict__ dst,
                           unsigned* __restrict__ fill_pos, unsigned* __restrict__ col_src,
                           int E) {
    int e = blockIdx.x * blockDim.x + threadIdx.x;
    if (e < E) {
        unsigned pos = atomicAdd(&fill_pos[dst[e]], 1u);
        col_src[pos] = (unsigned)src[e];
    }
}

// ---------------------------------------------------------------------------
// Xs[i][:] = X[i][:] * dout[i]  (source-norm applied once per row)
// ---------------------------------------------------------------------------
__global__ void prescale(const float* __restrict__ X, const float* __restrict__ dout,
                         float* __restrict__ Xs, long nvec4) {
    long i = (long)blockIdx.x * blockDim.x + threadIdx.x;  // over N*32 float4s
    if (i < nvec4) {
        int row = (int)(i >> 5);
        float s = dout[row];
        float4 v = ((const float4*)X)[i];
        v.x *= s; v.y *= s; v.z *= s; v.w *= s;
        ((float4*)Xs)[i] = v;
    }
}

// ---------------------------------------------------------------------------
// CSR gather SpMM: one wave per dst node, lane = float4 of features.
// acc starts at Xs[node] (self-loop); every edge is a coalesced 512B row read.
// No atomics, deterministic accumulation order.
// ---------------------------------------------------------------------------
__global__ void spmm_gather(const float* __restrict__ Xs, const unsigned* __restrict__ row_ptr,
                            const unsigned* __restrict__ col_src, float* __restrict__ agg,
                            int N) {
    int node = (blockIdx.x << 3) + (threadIdx.x >> 5);  // 8 nodes per 256-thread block
    if (node >= N) return;
    int lane = threadIdx.x & 31;
    float4 acc = ((const float4*)(Xs + (size_t)node * FDIM))[lane];  // self-loop
    unsigned beg = row_ptr[node];
    unsigned end = row_ptr[node + 1];
    for (unsigned i = beg; i < end; ++i) {
        unsigned s = col_src[i];
        if (i + 1u < end)
            __builtin_prefetch(Xs + (size_t)col_src[i + 1u] * FDIM + (lane << 2), 0, 0);
        float4 v = ((const float4*)(Xs + (size_t)s * FDIM))[lane];
        acc.x += v.x; acc.y += v.y; acc.z += v.z; acc.w += v.w;
    }
    ((float4*)(agg + (size_t)node * FDIM))[lane] = acc;
}

// ---------------------------------------------------------------------------
// Pack W (128 x FOUT) into k-pair-major float2: Wp[p*FOUT+c] = {W[2p][c], W[2p+1][c]}
// so each lane's B-fragment per k-step is a single b64 load.
// ---------------------------------------------------------------------------
__global__ void repack_w(const float* __restrict__ W, float* __restrict__ Wp, int fout) {
    int idx = blockIdx.x * blockDim.x + threadIdx.x;  // over 64*fout pairs
    int total = (FDIM / 2) * fout;
    if (idx < total) {
        int p = idx / fout;
        int c = idx - p * fout;
        Wp[2 * idx + 0] = W[(size_t)(2 * p) * fout + c];
        Wp[2 * idx + 1] = W[(size_t)(2 * p + 1) * fout + c];
    }
}

// ---------------------------------------------------------------------------
// Y = relu(din[row] * (agg @ W) + b) via V_WMMA_F32_16X16X4_F32 (full fp32).
// din folded into the epilogue, so the A-tile stage is a pure copy -> async
// global->LDS b128 when the toolchain exposes it. A padded to 132 floats/row
// (conflict-free strided reads). Safe to run in place (Y == agg): the tile is
// fully staged to LDS + barrier before any store, per-block rows disjoint.
// ---------------------------------------------------------------------------
template <int FOUT>
__global__ void gemm_bias_relu(const float* agg, const float* __restrict__ din,
                               const float* __restrict__ Wp, const float* __restrict__ bias,
                               float* Y, int nrows) {
    __shared__ float lds_a[16 * 132];
    __shared__ float lds_din[16];

    const int r0 = blockIdx.x * 16;

    if (threadIdx.x < 16) {
        int gr = r0 + (int)threadIdx.x;
        if (gr >= nrows) gr = nrows - 1;
        lds_din[threadIdx.x] = din[gr];
    }

    // Stage 16x128 A tile (pure copy), 512 float4s.
    for (int idx = threadIdx.x; idx < 16 * 32; idx += blockDim.x) {
        int row = idx >> 5;
        int c4  = idx & 31;
        int gr = r0 + row;
        if (gr >= nrows) gr = nrows - 1;
#if HAVE_ASYNC
        __builtin_amdgcn_global_load_async_to_lds_b128(
            (__attribute__((address_space(1))) v4i*)(agg + (size_t)gr * FDIM + (c4 << 2)),
            (__attribute__((address_space(3))) v4i*)&lds_a[row * 132 + (c4 << 2)],
            0, 0);
#else
        float4 v = ((const float4*)(agg + (size_t)gr * FDIM))[c4];
        *((float4*)&lds_a[row * 132 + (c4 << 2)]) = v;
#endif
    }
#if HAVE_ASYNC
#if __has_builtin(__builtin_amdgcn_s_wait_asynccnt)
    __builtin_amdgcn_s_wait_asynccnt(0);
#else
    asm volatile("s_wait_asynccnt 0x0" ::: "memory");
#endif
#endif
    __syncthreads();

    const int wave = threadIdx.x >> 5;
    const int lane = threadIdx.x & 31;
    const int half = lane >> 4;   // 0: K=2p, 1: K=2p+1 pair selection
    const int l    = lane & 15;
    const int col0 = wave * 16;

    const float bv = bias[col0 + l];
    const v2f* Wpv = (const v2f*)Wp;

    v8f c = {0.f, 0.f, 0.f, 0.f, 0.f, 0.f, 0.f, 0.f};

#pragma unroll
    for (int k = 0; k < FDIM; k += 4) {
        v2f a;
        a.x = lds_a[l * 132 + k + 2 * half];
        a.y = lds_a[l * 132 + k + 2 * half + 1];
        v2f b = Wpv[(size_t)((k >> 1) + half) * FOUT + col0 + l];
        c = __builtin_amdgcn_wmma_f32_16x16x4_f32(false, a, false, b, (short)0, c,
                                                  false, false);
    }

    // C/D layout: VGPR j -> M = j + 8*half, N = l
#pragma unroll
    for (int j = 0; j < 8; ++j) {
        int row = r0 + j + 8 * half;
        if (row < nrows) {
            float v = c[j] * lds_din[j + 8 * half] + bv;
            Y[(size_t)row * FOUT + col0 + l] = v > 0.f ? v : 0.f;
        }
    }
}

// ---------------------------------------------------------------------------
// ws layout (4B units):
//   dout[N] | din[N] | Xs[N*128] | agg[N*128] | row_ptr[N+1] | fill_pos[N]
//   | col_src[E] | Wp1[16384] | Wp2[16384] | Wp3[8192]
// ---------------------------------------------------------------------------
extern "C" void kernel_launch(void* const* d_in, const int* in_sizes, int n_in,
                              void* d_out, int out_size, void* d_ws, size_t ws_size,
                              hipStream_t stream) {
    const int*   src  = (const int*)d_in[0];
    const int*   dst  = (const int*)d_in[1];
    const float* feat = (const float*)d_in[2];
    const float* W1   = (const float*)d_in[3];
    const float* b1   = (const float*)d_in[4];
    const float* W2   = (const float*)d_in[5];
    const float* b2   = (const float*)d_in[6];
    const float* W3   = (const float*)d_in[7];
    const float* b3   = (const float*)d_in[8];

    const int E = in_sizes[0];
    const int N = in_sizes[2] / FDIM;

    float* ws   = (float*)d_ws;
    float* dout = ws;
    float* din  = ws + N;
    float* Xs   = din + N;
    float* agg  = Xs + (size_t)N * FDIM;
    unsigned* row_ptr  = (unsigned*)(agg + (size_t)N * FDIM);
    unsigned* fill_pos = row_ptr + (N + 1);
    unsigned* col_src  = fill_pos + N;
    float* Wp1 = (float*)(col_src + E);
    float* Wp2 = Wp1 + (FDIM / 2) * 128 * 2;
    float* Wp3 = Wp2 + (FDIM / 2) * 128 * 2;
    float* out = (float*)d_out;

    const long nvec4  = (long)N * (FDIM / 4);
    const int  nblkN  = (N + 255) / 256;
    const int  nblkE  = (E + 255) / 256;
    const int  nblkV4 = (int)((nvec4 + 255) / 256);
    const int  nblkSp = (N + 7) / 8;
    const int  nblkGm = (N + 15) / 16;

    // --- Graph preprocessing: degrees + CSR-by-destination ---
    deg_init<<<nblkN, 256, 0, stream>>>((unsigned*)dout, (unsigned*)din, N);
    deg_acc<<<nblkE, 256, 0, stream>>>(src, dst, (unsigned*)dout, (unsigned*)din, E);
    scan_rowptr<<<1, SCAN_T, 0, stream>>>((const unsigned*)din, row_ptr, N);
    copy_u32<<<nblkN, 256, 0, stream>>>(row_ptr, fill_pos, N);
    fill_edges<<<nblkE, 256, 0, stream>>>(src, dst, fill_pos, col_src, E);
    deg_fin<<<nblkN, 256, 0, stream>>>(dout, din, N);

    // --- Weight repack (k-pair-major float2) ---
    repack_w<<<(64 * 128 + 255) / 256, 256, 0, stream>>>(W1, Wp1, 128);
    repack_w<<<(64 * 128 + 255) / 256, 256, 0, stream>>>(W2, Wp2, 128);
    repack_w<<<(64 * 64 + 255) / 256, 256, 0, stream>>>(W3, Wp3, 64);

    // --- Layer 1: features -> agg (in-place GEMM) ---
    prescale<<<nblkV4, 256, 0, stream>>>(feat, dout, Xs, nvec4);
    spmm_gather<<<nblkSp, 256, 0, stream>>>(Xs, row_ptr, col_src, agg, N);
    gemm_bias_relu<128><<<nblkGm, 256, 0, stream>>>(agg, din, Wp1, b1, agg, N);

    // --- Layer 2: agg -> agg ---
    prescale<<<nblkV4, 256, 0, stream>>>(agg, dout, Xs, nvec4);
    spmm_gather<<<nblkSp, 256, 0, stream>>>(Xs, row_ptr, col_src, agg, N);
    gemm_bias_relu<128><<<nblkGm, 256, 0, stream>>>(agg, din, Wp2, b2, agg, N);

    // --- Layer 3: agg -> out (FOUT = 64, 4 waves/block) ---
    prescale<<<nblkV4, 256, 0, stream>>>(agg, dout, Xs, nvec4);
    spmm_gather<<<nblkSp, 256, 0, stream>>>(Xs, row_ptr, col_src, agg, N);
    gemm_bias_relu<64><<<nblkGm, 128, 0, stream>>>(agg, din, Wp3, b3, out, N);
}